// VQEmbeddingGSSoft_51419348467988
// MI455X (gfx1250) — compile-verified
//
#include <hip/hip_runtime.h>
#include <hip/hip_bf16.h>

// Problem constants (from the reference)
#define NN 8
#define MM 512
#define DD 64
#define BB 8192

// Output layout: quantized (B*N*D) | KL (1) | perplexity (1) | ret_logits (B*N*M)
static constexpr size_t Q_OFF  = 0;
static constexpr size_t KL_OFF = (size_t)BB * NN * DD;          // 4194304
static constexpr size_t PX_OFF = KL_OFF + 1;
static constexpr size_t LG_OFF = PX_OFF + 1;                    // 4194306

// Workspace layout (floats): esq[N*M] | avg[N*M] | kl[1]
static constexpr size_t WS_ESQ = 0;
static constexpr size_t WS_AVG = (size_t)NN * MM;
static constexpr size_t WS_KL  = WS_AVG + (size_t)NN * MM;

typedef __attribute__((ext_vector_type(2))) float v2f;
typedef __attribute__((ext_vector_type(8))) float v8f;

__device__ __forceinline__ float wred_max(float v) {
#pragma unroll
  for (int off = 16; off > 0; off >>= 1) v = fmaxf(v, __shfl_xor(v, off, 32));
  return v;
}
__device__ __forceinline__ float wred_sum(float v) {
#pragma unroll
  for (int off = 16; off > 0; off >>= 1) v += __shfl_xor(v, off, 32);
  return v;
}

// ---------------------------------------------------------------------------
// pass0: e_sq[n][m] = sum_d e[n][m][d]^2 ; zero avg_probs and KL accumulators
// ---------------------------------------------------------------------------
__global__ __launch_bounds__(256) void vq_pass0(const float* __restrict__ emb,
                                                float* __restrict__ ws) {
  int idx = blockIdx.x * 256 + threadIdx.x;
  if (idx == 0) ws[WS_KL] = 0.0f;
  if (idx < NN * MM) {
    const float* e = emb + (size_t)idx * DD;
    float s = 0.0f;
#pragma unroll 8
    for (int d = 0; d < DD; ++d) s += e[d] * e[d];
    ws[WS_ESQ + idx] = s;
    ws[WS_AVG + idx] = 0.0f;
  }
}

// ---------------------------------------------------------------------------
// pass1: raw logits s[b,m] = 2*x.e - |e|^2 via V_WMMA_F32_16X16X4_F32.
// Block = 4 waves, each wave owns 16 consecutive b-rows for one n.
// Writes s into the ret_logits region of d_out (finalized by pass2).
// ---------------------------------------------------------------------------
__global__ __launch_bounds__(128) void vq_pass1(const float* __restrict__ x,
                                                const float* __restrict__ emb,
                                                const float* __restrict__ ws,
                                                float* __restrict__ out) {
  const int n    = blockIdx.y;
  const int wave = threadIdx.x >> 5;
  const int lane = threadIdx.x & 31;
  const int b0   = blockIdx.x * 64 + wave * 16;
  const int r    = lane & 15;          // A row / B col / D col (lane%16)
  const int kh   = (lane >> 4) * 2;    // K sub-offset per lane half (0 or 2)
  const int rhi  = (lane >> 4) * 8;    // D row offset per lane half

  float* logits = out + LG_OFF;
  const float* esq = ws + WS_ESQ + (size_t)n * MM;

  // Preload A fragments: A[row=r][k], 16 chunks of K=4 (D=64 total)
  float a0[16], a1[16];
  const float* xrow = x + (size_t)(b0 + r) * (NN * DD) + (size_t)n * DD;
#pragma unroll
  for (int kc = 0; kc < 16; ++kc) {
    int k = kc * 4 + kh;
    a0[kc] = xrow[k];
    a1[kc] = xrow[k + 1];
  }

  const float* en = emb + (size_t)n * MM * DD;
  for (int m0 = 0; m0 < MM; m0 += 16) {
    v8f acc = {0.f, 0.f, 0.f, 0.f, 0.f, 0.f, 0.f, 0.f};
    // B fragment: B[k][j] = 2*e[m0+j][k], j = r, lane-half selects k pair
    const float* ecol = en + (size_t)(m0 + r) * DD;
#pragma unroll
    for (int kc = 0; kc < 16; ++kc) {
      int k = kc * 4 + kh;
      v2f A = {a0[kc], a1[kc]};
      v2f Bf = {2.0f * ecol[k], 2.0f * ecol[k + 1]};
      acc = __builtin_amdgcn_wmma_f32_16x16x4_f32(
          /*neg_a=*/false, A, /*neg_b=*/false, Bf,
          /*c_mod=*/(short)0, acc, /*reuse_a=*/false, /*reuse_b=*/false);
    }
    float es = esq[m0 + r];
#pragma unroll
    for (int vr = 0; vr < 8; ++vr) {
      int row = vr + rhi;  // D layout: row = vgpr + 8*(lane/16), col = lane%16
      logits[(size_t)(b0 + row) * (NN * MM) + (size_t)n * MM + m0 + r] =
          acc[vr] - es;
    }
  }
}

// ---------------------------------------------------------------------------
// pass2: per-row log-softmax (overwrite s -> log_probs), Gumbel sampling,
// KL/avg_probs accumulation, quantized = samples @ e.
// Block = 8 waves = 256 threads; each wave processes 8 rows sequentially.
// ---------------------------------------------------------------------------
__global__ __launch_bounds__(256) void vq_pass2(const float* __restrict__ emb,
                                                const float* __restrict__ u,
                                                const float* __restrict__ tau,
                                                float* __restrict__ out,
                                                float* __restrict__ ws) {
  const int n    = blockIdx.y;
  const int wave = threadIdx.x >> 5;
  const int lane = threadIdx.x & 31;
  const float LOGM = 6.238324625039508f;  // log(512)

  __shared__ float s_avg[MM];
  __shared__ float s_samp[8][MM];

  for (int i = threadIdx.x; i < MM; i += 256) s_avg[i] = 0.0f;
  __syncthreads();

  const float inv_tau = 1.0f / tau[0];
  float* logits = out + LG_OFF;
  const float* en = emb + (size_t)n * MM * DD;
  float kl_acc = 0.0f;

  for (int it = 0; it < 8; ++it) {
    const int b = blockIdx.x * 64 + wave * 8 + it;
    const size_t rowoff = (size_t)b * (NN * MM) + (size_t)n * MM;

    // --- stable log-softmax over M (16 values per lane) ---
    float sv[16];
    float mx = -3.4e38f;
#pragma unroll
    for (int i = 0; i < 16; ++i) {
      sv[i] = logits[rowoff + lane + 32 * i];
      mx = fmaxf(mx, sv[i]);
    }
    mx = wred_max(mx);
    float se = 0.0f;
#pragma unroll
    for (int i = 0; i < 16; ++i) se += __expf(sv[i] - mx);
    se = wred_sum(se);
    const float lse = mx + __logf(se);

    // --- log_probs writeback, KL partial, Gumbel logits ---
    const float* urow = u + ((size_t)n * BB + b) * MM;
    float yv[16];
    float ymax = -3.4e38f;
#pragma unroll
    for (int i = 0; i < 16; ++i) {
      float lp = sv[i] - lse;
      logits[rowoff + lane + 32 * i] = lp;
      float p = __expf(lp);
      kl_acc += p * (lp + LOGM);
      float uu = urow[lane + 32 * i];
      float g = -__logf(-__logf(uu));
      float y = (lp + g) * inv_tau;
      yv[i] = y;
      ymax = fmaxf(ymax, y);
    }
    ymax = wred_max(ymax);
    float ys = 0.0f;
#pragma unroll
    for (int i = 0; i < 16; ++i) {
      yv[i] = __expf(yv[i] - ymax);
      ys += yv[i];
    }
    ys = wred_sum(ys);
    const float inv_ys = 1.0f / ys;

    // --- samples -> LDS, avg_probs accumulate ---
#pragma unroll
    for (int i = 0; i < 16; ++i) {
      float samp = yv[i] * inv_ys;
      s_samp[wave][lane + 32 * i] = samp;
      atomicAdd(&s_avg[lane + 32 * i], samp);
    }
    __syncthreads();  // uniform across all 8 waves (same trip counts)

    // --- quantized[b][d] = sum_m samples[m] * e[n][m][d]; 2 d's per lane ---
    const int d0 = lane * 2;
    float q0 = 0.0f, q1 = 0.0f;
    for (int m = 0; m < MM; ++m) {
      float sm = s_samp[wave][m];
      q0 = fmaf(sm, en[(size_t)m * DD + d0], q0);
      q1 = fmaf(sm, en[(size_t)m * DD + d0 + 1], q1);
    }
    float* qo = out + Q_OFF + (size_t)b * (NN * DD) + (size_t)n * DD;
    qo[d0]     = q0;
    qo[d0 + 1] = q1;
    __syncthreads();
  }

  // --- KL reduction ---
  kl_acc = wred_sum(kl_acc);
  if (lane == 0) atomicAdd(ws + WS_KL, kl_acc);

  __syncthreads();
  for (int i = threadIdx.x; i < MM; i += 256)
    atomicAdd(ws + WS_AVG + (size_t)n * MM + i, s_avg[i]);
}

// ---------------------------------------------------------------------------
// pass3: scalars. KL = total/B; perplexity = sum_n exp(-sum_m ap*log(ap+1e-10))
// ---------------------------------------------------------------------------
__global__ __launch_bounds__(256) void vq_pass3(const float* __restrict__ ws,
                                                float* __restrict__ out) {
  __shared__ float red[256];
  const float invB = 1.0f / (float)BB;
  float acc = 0.0f;
  for (int n = 0; n < NN; ++n) {
    float part = 0.0f;
    for (int m = threadIdx.x; m < MM; m += 256) {
      float ap = ws[WS_AVG + (size_t)n * MM + m] * invB;
      part += ap * __logf(ap + 1e-10f);
    }
    red[threadIdx.x] = part;
    __syncthreads();
    for (int s = 128; s > 0; s >>= 1) {
      if (threadIdx.x < s) red[threadIdx.x] += red[threadIdx.x + s];
      __syncthreads();
    }
    if (threadIdx.x == 0) acc += __expf(-red[0]);
    __syncthreads();
  }
  if (threadIdx.x == 0) {
    out[PX_OFF] = acc;
    out[KL_OFF] = ws[WS_KL] * invB;
  }
}

// ---------------------------------------------------------------------------
extern "C" void kernel_launch(void* const* d_in, const int* in_sizes, int n_in,
                              void* d_out, int out_size, void* d_ws,
                              size_t ws_size, hipStream_t stream) {
  const float* x   = (const float*)d_in[0];
  const float* emb = (const float*)d_in[1];
  const float* u   = (const float*)d_in[2];
  const float* tau = (const float*)d_in[3];
  float* out = (float*)d_out;
  float* ws  = (float*)d_ws;

  // pass0: e_sq + zero accumulators (N*M = 4096 work items)
  vq_pass0<<<dim3((NN * MM + 255) / 256), dim3(256), 0, stream>>>(emb, ws);

  // pass1: WMMA distance GEMM; 64 rows/block, grid (B/64, N)
  vq_pass1<<<dim3(BB / 64, NN), dim3(128), 0, stream>>>(x, emb, ws, out);

  // pass2: softmax/Gumbel/quantize; 64 rows/block, grid (B/64, N)
  vq_pass2<<<dim3(BB / 64, NN), dim3(256), 0, stream>>>(emb, u, tau, out, ws);

  // pass3: scalars
  vq_pass3<<<dim3(1), dim3(256), 0, stream>>>(ws, out);
}